// RQSQuantile_48043504173067
// MI455X (gfx1250) — compile-verified
//
#include <hip/hip_runtime.h>
#include <stdint.h>

// ---------------------------------------------------------------------------
// RQS quantile transform for MI455X (gfx1250).
// Phase 1: build per-dim spline tables (k-major) in workspace.
// Phase 2: bandwidth-bound streaming kernel; spline tables staged into LDS
//          via the CDNA5 Tensor Data Mover (tensor_load_to_lds + TENSORcnt).
// ---------------------------------------------------------------------------

#define BOUNDV    10.0f
#define EPSV      1e-6f
#define MIN_SCALE 1e-4f
#define KBINS     64      // K (fixed by the problem)
#define NKNOT     65      // K+1
#define CTILE     64      // columns per block
#define RPB       256     // rows per block
#define THREADS   256

typedef __attribute__((ext_vector_type(4))) unsigned int v4u_t;
typedef __attribute__((ext_vector_type(8))) int          v8i_t;
typedef __attribute__((ext_vector_type(4))) int          v4i_t;

#if defined(__gfx1250__) && __has_builtin(__builtin_amdgcn_tensor_load_to_lds)
#define HAS_TDM 1
#else
#define HAS_TDM 0
#endif

__device__ __forceinline__ float softplusf(float x) {
  // numerically stable log1p(exp(x))
  return fmaxf(x, 0.0f) + log1pf(__expf(-fabsf(x)));
}

// ---------------------------------------------------------------------------
// Phase 1: per-dimension spline tables, written k-major: tbl[k*D + d].
// D=1024 threads total; each thread handles one dimension serially (tiny).
// ---------------------------------------------------------------------------
__global__ void rqs_setup_kernel(const float* __restrict__ log_scale,
                                 const float* __restrict__ bias,
                                 const float* __restrict__ raw_w,
                                 const float* __restrict__ raw_h,
                                 const float* __restrict__ raw_s,
                                 float* __restrict__ xk_t,   // [NKNOT][D]
                                 float* __restrict__ yk_t,   // [NKNOT][D]
                                 float* __restrict__ sl_t,   // [NKNOT][D]
                                 float* __restrict__ scale_t,// [D]
                                 float* __restrict__ bias_t, // [D]
                                 int D) {
  int d = blockIdx.x * blockDim.x + threadIdx.x;
  if (d >= D) return;
  const float twob = 2.0f * BOUNDV;

  {  // softmax(raw_w) * 2B -> cumsum -> x-knots
    const float* w = raw_w + (size_t)d * KBINS;
    float m = -3.4e38f;
    for (int k = 0; k < KBINS; ++k) m = fmaxf(m, w[k]);
    float s = 0.0f;
    for (int k = 0; k < KBINS; ++k) s += __expf(w[k] - m);
    float inv = twob / s;
    float acc = -BOUNDV;
    xk_t[d] = acc;
    for (int k = 0; k < KBINS; ++k) {
      acc += __expf(w[k] - m) * inv;
      xk_t[(size_t)(k + 1) * D + d] = acc;
    }
  }
  {  // softmax(raw_h) * 2B -> cumsum -> y-knots
    const float* h = raw_h + (size_t)d * KBINS;
    float m = -3.4e38f;
    for (int k = 0; k < KBINS; ++k) m = fmaxf(m, h[k]);
    float s = 0.0f;
    for (int k = 0; k < KBINS; ++k) s += __expf(h[k] - m);
    float inv = twob / s;
    float acc = -BOUNDV;
    yk_t[d] = acc;
    for (int k = 0; k < KBINS; ++k) {
      acc += __expf(h[k] - m) * inv;
      yk_t[(size_t)(k + 1) * D + d] = acc;
    }
  }
  {  // softplus slopes
    const float* sv = raw_s + (size_t)d * NKNOT;
    for (int k = 0; k < NKNOT; ++k)
      sl_t[(size_t)k * D + d] = softplusf(sv[k]);
  }
  scale_t[d] = softplusf(log_scale[d]) + MIN_SCALE;
  bias_t[d]  = bias[d];
}

// ---------------------------------------------------------------------------
// TDM 2D tile load: tile_d0 contiguous elements x tile_d1 rows, row stride
// `stride0` (all in 4-byte element units), packed densely into LDS.
// Descriptor layout per CDNA5 ISA.
// This toolchain exposes the 6-arg builtin:
//   (uint32x4 g0, int32x8 g1, int32x4 g2, int32x4 g3, int32x8 g4, i32 cpol)
// ---------------------------------------------------------------------------
#if HAS_TDM
__device__ __forceinline__ void tdm_load_tile_2d(uint32_t lds_addr,
                                                 const float* gptr,
                                                 uint32_t tensor_d0,
                                                 uint32_t tensor_d1,
                                                 uint32_t tile_d0,
                                                 uint32_t tile_d1,
                                                 uint32_t stride0) {
  uint64_t ga = (uint64_t)(uintptr_t)gptr;
  v4u_t g0 = {0u, 0u, 0u, 0u};
  g0[0] = 1u;                                   // count=1, user descriptor
  g0[1] = lds_addr;                             // LDS byte address
  g0[2] = (uint32_t)ga;                         // global_addr[31:0]
  g0[3] = (uint32_t)((ga >> 32) & 0x1FFFFFFu)   // global_addr[56:32]
        | (2u << 30);                           // type = 2 ("image")
  v8i_t g1 = {0, 0, 0, 0, 0, 0, 0, 0};
  g1[0] = (int)(2u << 16);                      // data_size=2 (4 bytes), mask=0
  g1[1] = (int)((tensor_d0 & 0xFFFFu) << 16);   // tensor_dim0[15:0]
  g1[2] = (int)(((tensor_d0 >> 16) & 0xFFFFu)   // tensor_dim0[31:16]
        | ((tensor_d1 & 0xFFFFu) << 16));       // tensor_dim1[15:0]
  g1[3] = (int)(((tensor_d1 >> 16) & 0xFFFFu)   // tensor_dim1[31:16]
        | ((tile_d0 & 0xFFFFu) << 16));         // tile_dim0
  g1[4] = (int)(tile_d1 & 0xFFFFu);             // tile_dim1 (tile_dim2=0)
  g1[5] = (int)stride0;                         // tensor_dim0_stride[31:0]
  g1[6] = 0;                                    // stride0 hi / stride1 lo
  g1[7] = 0;
  v4i_t z4 = {0, 0, 0, 0};                      // 2D tensor: groups 2/3 unused
  v8i_t z8 = {0, 0, 0, 0, 0, 0, 0, 0};          // extra group (unused)
  __builtin_amdgcn_tensor_load_to_lds(g0, g1, z4, z4, z8, 0);
}
#endif

// ---------------------------------------------------------------------------
// Phase 2: main streaming kernel.
// Block = 256 threads, owns CTILE=64 columns x RPB=256 rows.
// Thread mapping: (threadIdx.x & 15) -> 4-column group (float4 elements),
//                 (threadIdx.x >> 4) -> row lane; 16 rows per iteration.
// ---------------------------------------------------------------------------
__global__ __launch_bounds__(THREADS)
void rqs_main_kernel(const float* __restrict__ u,
                     const float* __restrict__ tau,
                     const float* __restrict__ xk_t,
                     const float* __restrict__ yk_t,
                     const float* __restrict__ sl_t,
                     const float* __restrict__ scale_t,
                     const float* __restrict__ bias_t,
                     float* __restrict__ out,
                     int B, int D) {
  __shared__ float sX[NKNOT * CTILE];
  __shared__ float sY[NKNOT * CTILE];
  __shared__ float sS[NKNOT * CTILE];
  __shared__ float sScale[CTILE];
  __shared__ float sBias[CTILE];

  const int c0 = blockIdx.x * CTILE;
  const int r0 = blockIdx.y * RPB;

  // small per-column affine params via plain loads
  for (int i = threadIdx.x; i < CTILE; i += THREADS) {
    sScale[i] = scale_t[c0 + i];
    sBias[i]  = bias_t[c0 + i];
  }

#if HAS_TDM
  if (threadIdx.x < 32) {  // wave 0 issues the 3 DMA descriptors once
    tdm_load_tile_2d((uint32_t)(uintptr_t)(void*)sX, xk_t + c0,
                     (uint32_t)D, NKNOT, CTILE, NKNOT, (uint32_t)D);
    tdm_load_tile_2d((uint32_t)(uintptr_t)(void*)sY, yk_t + c0,
                     (uint32_t)D, NKNOT, CTILE, NKNOT, (uint32_t)D);
    tdm_load_tile_2d((uint32_t)(uintptr_t)(void*)sS, sl_t + c0,
                     (uint32_t)D, NKNOT, CTILE, NKNOT, (uint32_t)D);
    __builtin_amdgcn_s_wait_tensorcnt(0);
  }
#else
  for (int i = threadIdx.x; i < NKNOT * CTILE; i += THREADS) {
    int k = i / CTILE, c = i % CTILE;
    size_t g = (size_t)k * D + c0 + c;
    sX[i] = xk_t[g];
    sY[i] = yk_t[g];
    sS[i] = sl_t[g];
  }
#endif
  __syncthreads();

  const int cg  = threadIdx.x & 15;   // column group (4 cols each)
  const int rl  = threadIdx.x >> 4;   // row lane 0..15
  const int clb = cg * 4;             // local column base
  const int rEnd = (r0 + RPB < B) ? (r0 + RPB) : B;

  for (int r = r0 + rl; r < rEnd; r += 16) {
    const float tv = tau[r];
    const float4 uv =
        *(const float4*)(u + (size_t)r * D + (size_t)(c0 + clb));
    float res[4];
    const float uin[4] = {uv.x, uv.y, uv.z, uv.w};
#pragma unroll
    for (int j = 0; j < 4; ++j) {
      const int c = clb + j;
      // logit with clipping
      float us = fminf(fmaxf(uin[j], EPSV), 1.0f - EPSV);
      float z  = __logf(us) - __logf(1.0f - us);
      float zc = fminf(fmaxf(z, -BOUNDV), BOUNDV);
      // 6-step binary search over 64 bins in LDS (K=64 fixed)
      int lo = 0;
#pragma unroll
      for (int st = 32; st >= 1; st >>= 1) {
        int m = lo + st;                 // probes <= 63
        lo = (sX[m * CTILE + c] <= zc) ? m : lo;
      }
      const float x0 = sX[lo * CTILE + c];
      const float x1 = sX[(lo + 1) * CTILE + c];
      const float y0 = sY[lo * CTILE + c];
      const float y1 = sY[(lo + 1) * CTILE + c];
      const float d0 = sS[lo * CTILE + c];
      const float d1 = sS[(lo + 1) * CTILE + c];
      const float w    = x1 - x0;
      const float h    = y1 - y0;
      const float invw = __builtin_amdgcn_rcpf(w);
      const float th   = (zc - x0) * invw;
      const float sl   = h * invw;
      const float t1m  = th * (1.0f - th);
      const float den  = sl + (d0 + d1 - 2.0f * sl) * t1m;
      const float yv   = y0 + h * (sl * th * th + d0 * t1m)
                                 * __builtin_amdgcn_rcpf(den);
      const bool inside = (z > -BOUNDV) && (z < BOUNDV);
      const float rq = inside ? yv : z;
      res[j] = tv * (rq * sScale[c] + sBias[c]);
    }
    float4 ov = make_float4(res[0], res[1], res[2], res[3]);
    *(float4*)(out + (size_t)r * D + (size_t)(c0 + clb)) = ov;
  }
}

// ---------------------------------------------------------------------------
// Host launcher. Inputs: u, tau, log_scale, bias, raw_w, raw_h, raw_s.
// Workspace layout (floats): xk_t | yk_t | sl_t | scale_t | bias_t
//   = 3*(K+1)*D + 2*D floats  (~807 KB for D=1024, K=64).
// ---------------------------------------------------------------------------
extern "C" void kernel_launch(void* const* d_in, const int* in_sizes, int n_in,
                              void* d_out, int out_size, void* d_ws, size_t ws_size,
                              hipStream_t stream) {
  const float* u         = (const float*)d_in[0];
  const float* tau       = (const float*)d_in[1];
  const float* log_scale = (const float*)d_in[2];
  const float* bias      = (const float*)d_in[3];
  const float* raw_w     = (const float*)d_in[4];
  const float* raw_h     = (const float*)d_in[5];
  const float* raw_s     = (const float*)d_in[6];
  float* out = (float*)d_out;

  const int D = in_sizes[2];            // 1024
  const int B = in_sizes[0] / D;        // 8192
  // K is fixed at 64 by the problem (kernel binary search is hardwired).

  float* ws      = (float*)d_ws;
  float* xk_t    = ws;
  float* yk_t    = xk_t + (size_t)NKNOT * D;
  float* sl_t    = yk_t + (size_t)NKNOT * D;
  float* scale_t = sl_t + (size_t)NKNOT * D;
  float* bias_t  = scale_t + D;

  // Phase 1: spline tables (k-major, TDM-friendly layout)
  rqs_setup_kernel<<<(D + 255) / 256, 256, 0, stream>>>(
      log_scale, bias, raw_w, raw_h, raw_s,
      xk_t, yk_t, sl_t, scale_t, bias_t, D);

  // Phase 2: streaming transform
  dim3 grid(D / CTILE, (B + RPB - 1) / RPB);   // 16 x 32 = 512 blocks
  rqs_main_kernel<<<grid, THREADS, 0, stream>>>(
      u, tau, xk_t, yk_t, sl_t, scale_t, bias_t, out, B, D);
}